// QuantumAttention_5703716569525
// MI455X (gfx1250) — compile-verified
//
#include <hip/hip_runtime.h>

// ---------------------------------------------------------------------------
// QuantumAttention on MI455X (gfx1250, wave32, WMMA).
// All matrix math runs through V_WMMA_F32_16X16X4_F32 (exact fp32 path).
// Attention is flash-fused (no 536MB score materialization).
// ---------------------------------------------------------------------------

typedef __attribute__((ext_vector_type(2))) float v2f;
typedef __attribute__((ext_vector_type(8))) float v8f;

#define BB   2
#define SS   2048
#define HID  1024
#define NHD  16
#define QD   32
#define HD   64
#define MM   (BB * SS)      // 4096 rows

__device__ __forceinline__ v8f wmma4(v2f a, v2f b, v8f c) {
  // D = A(16x4 f32) * B(4x16 f32) + C(16x16 f32)
  return __builtin_amdgcn_wmma_f32_16x16x4_f32(
      false, a, false, b, (short)0, c, false, false);
}

__device__ __forceinline__ v8f zero8() {
  v8f z = {0.f, 0.f, 0.f, 0.f, 0.f, 0.f, 0.f, 0.f};
  return z;
}

// ---------------------------------------------------------------------------
// Generic GEMM + bias: C[M,N] = A[M,K] @ W[K,N] + bias[N]
// Block tile 64x64, BK=16. 128 threads = 4 waves; each wave owns 32x32
// (2x2 tiles of 16x16). LDS staging with conflict-free padding.
// ---------------------------------------------------------------------------
__global__ __launch_bounds__(128) void gemm_bias_wmma(
    const float* __restrict__ A, const float* __restrict__ W,
    const float* __restrict__ bias, float* __restrict__ C,
    int M, int N, int K)
{
  __shared__ float As[64][20];   // 64x16 + pad4 (row = 80B, 16B aligned)
  __shared__ float Bs[16][68];   // 16x64 + pad4 (row = 272B, 16B aligned)

  const int tid  = threadIdx.x;
  const int lane = tid & 31;
  const int w    = tid >> 5;
  const int l16  = lane & 15;
  const int h    = lane >> 4;

  const int m0 = blockIdx.y * 64;
  const int n0 = blockIdx.x * 64;
  const int wm = (w >> 1) * 32;
  const int wn = (w & 1) * 32;

  v8f acc[2][2];
  for (int i = 0; i < 2; ++i)
    for (int j = 0; j < 2; ++j)
      acc[i][j] = zero8();

  for (int kb = 0; kb < K; kb += 16) {
    __syncthreads();
    // stage A chunk 64x16 (1024 floats), float4 x 128 thr x 2 iters
    {
      const int idx = tid * 4;
      for (int it = 0; it < 2; ++it) {
        int id = idx + it * 512;
        int r = id >> 4, c = id & 15;
        *(float4*)&As[r][c] =
            *(const float4*)&A[(size_t)(m0 + r) * K + kb + c];
      }
      // stage B chunk 16x64 (1024 floats)
      for (int it = 0; it < 2; ++it) {
        int id = idx + it * 512;
        int r = id >> 6, c = id & 63;
        *(float4*)&Bs[r][c] =
            *(const float4*)&W[(size_t)(kb + r) * N + n0 + c];
      }
    }
    __syncthreads();

    for (int kk = 0; kk < 16; kk += 4) {
      v2f afrag[2], bfrag[2];
      for (int i = 0; i < 2; ++i)
        afrag[i] = *(const v2f*)&As[wm + i * 16 + l16][kk + 2 * h];
      for (int j = 0; j < 2; ++j) {
        const int col = wn + j * 16 + l16;
        v2f bv2;
        bv2.x = Bs[kk + 2 * h][col];
        bv2.y = Bs[kk + 2 * h + 1][col];
        bfrag[j] = bv2;
      }
      for (int i = 0; i < 2; ++i)
        for (int j = 0; j < 2; ++j)
          acc[i][j] = wmma4(afrag[i], bfrag[j], acc[i][j]);
    }
  }

  // epilogue: bias add + store (C/D layout: col=lane%16, row=r+8h)
  for (int j = 0; j < 2; ++j) {
    const int col = n0 + wn + j * 16 + l16;
    const float bv_ = bias[col];
    for (int i = 0; i < 2; ++i)
      for (int r = 0; r < 8; ++r) {
        const int row = m0 + wm + i * 16 + r + 8 * h;
        C[(size_t)row * N + col] = acc[i][j][r] + bv_;
      }
  }
}

// ---------------------------------------------------------------------------
// Row L2-normalize over 32 contiguous floats (one head vector per wave),
// with optional fused scale (1/sqrt(QD) folded into Q).
// ---------------------------------------------------------------------------
__global__ __launch_bounds__(256) void rownorm32(
    float* __restrict__ q, int nrows, float scale)
{
  const int wid  = (int)((blockIdx.x * blockDim.x + threadIdx.x) >> 5);
  const int lane = threadIdx.x & 31;
  if (wid >= nrows) return;
  const size_t off = (size_t)wid * 32 + lane;
  const float x = q[off];
  float ss = x * x;
  for (int m = 1; m < 32; m <<= 1) ss += __shfl_xor(ss, m, 32);
  const float nrm = fmaxf(sqrtf(ss), 1e-12f);
  q[off] = x / nrm * scale;
}

// ---------------------------------------------------------------------------
// Flash attention, fp32 WMMA core.
// grid = (S/64, NH, B), block = 128 (4 waves); wave owns 16 query rows.
// Per 64-key chunk: QK^T = 32 wmma, online softmax, P via LDS relayout,
// PV = 64 wmma.
// ---------------------------------------------------------------------------
__global__ __launch_bounds__(128) void flash_attn_wmma(
    const float* __restrict__ Qn, const float* __restrict__ Kn,
    const float* __restrict__ V,  const int* __restrict__ amask,
    float* __restrict__ Ctx)
{
  __shared__ float Ks[64][36];      // 64 keys x 32 (+pad4)
  __shared__ float Vs[64][68];      // 64 keys x 64 (+pad4)
  __shared__ float Ps[4][16][68];   // per-wave P tile 16q x 64k (+pad4)

  const int tid  = threadIdx.x;
  const int lane = tid & 31;
  const int w    = tid >> 5;
  const int l16  = lane & 15;
  const int h    = lane >> 4;

  const int b    = blockIdx.z;
  const int head = blockIdx.y;
  const int q0   = blockIdx.x * 64;
  const int qw   = q0 + w * 16;
  const size_t base = (size_t)b * SS;

  // Q fragments held in registers (A-layout), scale already folded in.
  v2f aQ[8];
  {
    const float* qrow = &Qn[(base + qw + l16) * (NHD * QD) + head * QD];
    for (int j = 0; j < 8; ++j)
      aQ[j] = *(const v2f*)&qrow[4 * j + 2 * h];
  }

  v8f O[4];
  for (int n = 0; n < 4; ++n) O[n] = zero8();
  float mrow[8], lrow[8];
  for (int r = 0; r < 8; ++r) { mrow[r] = -3.0e38f; lrow[r] = 0.f; }

  for (int kv0 = 0; kv0 < SS; kv0 += 64) {
    __syncthreads();  // previous chunk's LDS consumers done
    {
      const int idx = tid * 4;
      // stage K chunk 64x32 (2048 floats)
      for (int it = 0; it < 4; ++it) {
        int id = idx + it * 512;
        int r = id >> 5, c = id & 31;
        *(float4*)&Ks[r][c] =
            *(const float4*)&Kn[(base + kv0 + r) * (NHD * QD) + head * QD + c];
      }
      // stage V chunk 64x64 (4096 floats)
      for (int it = 0; it < 8; ++it) {
        int id = idx + it * 512;
        int r = id >> 6, c = id & 63;
        *(float4*)&Vs[r][c] =
            *(const float4*)&V[(base + kv0 + r) * HID + head * HD + c];
      }
    }
    __syncthreads();

    // --- scores: 4 tiles of 16q x 16k, K-dim 32 => 8 wmma each ---
    v8f Sv[4];
    int mk[4];
    for (int t = 0; t < 4; ++t)
      mk[t] = amask[base + kv0 + t * 16 + l16];

    for (int t = 0; t < 4; ++t) {
      v8f s = zero8();
      const float* krow = &Ks[t * 16 + l16][0];
      for (int j = 0; j < 8; ++j) {
        v2f bk2 = *(const v2f*)&krow[4 * j + 2 * h];
        s = wmma4(aQ[j], bk2, s);
      }
      if (mk[t] == 0)
        for (int r = 0; r < 8; ++r) s[r] = -3.0e38f;
      Sv[t] = s;
    }

    // --- online softmax row stats (row = r+8h; 16 cols live in the 16
    //     lanes sharing h, so xor masks 1..8 reduce within the half) ---
    float mnew[8], alpha[8], rsum[8];
    for (int r = 0; r < 8; ++r) {
      float v = fmaxf(fmaxf(Sv[0][r], Sv[1][r]), fmaxf(Sv[2][r], Sv[3][r]));
      for (int m = 1; m < 16; m <<= 1) v = fmaxf(v, __shfl_xor(v, m, 32));
      mnew[r]  = fmaxf(mrow[r], v);
      alpha[r] = __expf(mrow[r] - mnew[r]);
      rsum[r]  = 0.f;
    }

    // probabilities -> LDS (C-layout write), accumulate row sums
    for (int t = 0; t < 4; ++t)
      for (int r = 0; r < 8; ++r) {
        const float p = __expf(Sv[t][r] - mnew[r]);
        Ps[w][r + 8 * h][t * 16 + l16] = p;
        rsum[r] += p;
      }
    for (int r = 0; r < 8; ++r) {
      float v = rsum[r];
      for (int m = 1; m < 16; m <<= 1) v += __shfl_xor(v, m, 32);
      lrow[r] = lrow[r] * alpha[r] + v;
      mrow[r] = mnew[r];
    }

    // rescale accumulated O
    for (int n = 0; n < 4; ++n)
      for (int r = 0; r < 8; ++r) O[n][r] *= alpha[r];

    // make this wave's P stores visible before A-layout re-read
    asm volatile("s_wait_dscnt 0" ::: "memory");

    // --- PV: P(16x64) @ V(64x64), 16 K-steps x 4 N-tiles = 64 wmma ---
    for (int kk16 = 0; kk16 < 16; ++kk16) {
      const int kk = kk16 * 4;
      v2f aP = *(const v2f*)&Ps[w][l16][kk + 2 * h];
      for (int n = 0; n < 4; ++n) {
        v2f bv2;
        bv2.x = Vs[kk + 2 * h][n * 16 + l16];
        bv2.y = Vs[kk + 2 * h + 1][n * 16 + l16];
        O[n] = wmma4(aP, bv2, O[n]);
      }
    }
  }

  // write context (divide by softmax denominator)
  for (int n = 0; n < 4; ++n) {
    const int col = head * HD + n * 16 + l16;
    for (int r = 0; r < 8; ++r) {
      const int row = qw + r + 8 * h;
      Ctx[(base + row) * HID + col] = O[n][r] / lrow[r];
    }
  }
}

// ---------------------------------------------------------------------------
extern "C" void kernel_launch(void* const* d_in, const int* in_sizes, int n_in,
                              void* d_out, int out_size, void* d_ws, size_t ws_size,
                              hipStream_t stream) {
  const float* HS   = (const float*)d_in[0];
  const int*   mask = (const int*)  d_in[1];
  const float* Wq   = (const float*)d_in[2];
  const float* bq   = (const float*)d_in[3];
  const float* Wk   = (const float*)d_in[4];
  const float* bk   = (const float*)d_in[5];
  const float* Wv   = (const float*)d_in[6];
  const float* bv   = (const float*)d_in[7];
  const float* Wo   = (const float*)d_in[8];
  const float* bo   = (const float*)d_in[9];
  float* out = (float*)d_out;

  // workspace layout (floats): Q 8MB | K 8MB | V 16MB | Ctx 16MB = 48MB
  float* ws  = (float*)d_ws;
  float* Q   = ws;
  float* K   = ws + (size_t)MM * 512;            // +2,097,152
  float* Vb  = ws + (size_t)2 * MM * 512;        // +4,194,304
  float* Ctx = Vb + (size_t)MM * 1024;           // +8,388,608

  const dim3 blk(128);

  // projections
  gemm_bias_wmma<<<dim3(512 / 64,  MM / 64), blk, 0, stream>>>(HS, Wq, bq, Q,  MM, 512,  HID);
  gemm_bias_wmma<<<dim3(512 / 64,  MM / 64), blk, 0, stream>>>(HS, Wk, bk, K,  MM, 512,  HID);
  gemm_bias_wmma<<<dim3(1024 / 64, MM / 64), blk, 0, stream>>>(HS, Wv, bv, Vb, MM, 1024, HID);

  // quantum state prep: L2-normalize per (token, head); fold 1/sqrt(QD) into Q
  const int nrows = MM * NHD;                    // 65536 rows of 32
  rownorm32<<<dim3((nrows * 32 + 255) / 256), dim3(256), 0, stream>>>(Q, nrows, 0.1767766953f);
  rownorm32<<<dim3((nrows * 32 + 255) / 256), dim3(256), 0, stream>>>(K, nrows, 1.0f);

  // fused flash attention
  flash_attn_wmma<<<dim3(SS / 64, NHD, BB), blk, 0, stream>>>(Q, K, Vb, mask, Ctx);

  // output projection
  gemm_bias_wmma<<<dim3(1024 / 64, MM / 64), blk, 0, stream>>>(Ctx, Wo, bo, out, MM, 1024, HID);
}